// SphConv_55027120996903
// MI455X (gfx1250) — compile-verified
//
#include <hip/hip_runtime.h>
#include <math.h>

typedef float v2f __attribute__((ext_vector_type(2)));
typedef float v8f __attribute__((ext_vector_type(8)));

#define NGRID 32
#define LMAX  16
#define NM    31            // 2L-1 longitudinal modes
#define CIN   128
#define COUT  128
#define BATCH 64
#define KJP   1024          // n*n (theta x phi)
#define KLM   512           // LMAX * 32 (m padded 31->32)

// workspace layout (float element offsets)
#define WS_Q      0                           // 16*31*32 = 15872
#define WS_A1RE   16384                       // 512*1024
#define WS_A1IM   (WS_A1RE + KLM*KJP)
#define WS_A2RE   (WS_A1IM + KLM*KJP)
#define WS_A2IM   (WS_A2RE + KJP*KLM)
#define WS_COEFRE (WS_A2IM + KJP*KLM)         // 64*512*128
#define WS_COEFIM (WS_COEFRE + BATCH*KLM*CIN)
#define WS_CHATRE (WS_COEFIM + BATCH*KLM*CIN)
#define WS_CHATIM (WS_CHATRE + BATCH*KLM*COUT)

#define PI_D 3.14159265358979323846

// ---------------------------------------------------------------------------
// Kernel 0: normalized associated Legendre Q[l][m_idx][j] (m_idx = m_hat+15)
// One thread per theta sample j; recurrences kept per-lane in LDS.
// ---------------------------------------------------------------------------
__global__ void sph_legendre_kernel(float* __restrict__ ws) {
  __shared__ float P[LMAX][LMAX][NGRID];   // 32 KB
  const int j = threadIdx.x;               // 0..31
  const double theta = PI_D * (j + 0.5) / NGRID;
  const double x = cos(theta), sx = sin(theta);

  P[0][0][j] = 1.0f;
  for (int m = 1; m < LMAX; ++m)
    P[m][m][j] = (float)(-(2.0 * m - 1.0) * sx) * P[m - 1][m - 1][j];
  for (int m = 0; m < LMAX - 1; ++m) {
    P[m + 1][m][j] = (float)((2.0 * m + 1.0) * x) * P[m][m][j];
    for (int l = m + 2; l < LMAX; ++l)
      P[l][m][j] = ((float)((2.0 * l - 1.0) * x) * P[l - 1][m][j] -
                    (float)(l + m - 1) * P[l - 2][m][j]) / (float)(l - m);
  }

  float* Q = ws + WS_Q;
  for (int l = 0; l < LMAX; ++l) {
    for (int m = 0; m <= l; ++m) {
      double ratio = 1.0;                          // (l-m)!/(l+m)!
      for (int t = l - m + 1; t <= l + m; ++t) ratio /= (double)t;
      const float Nlm = (float)sqrt((2.0 * l + 1.0) / (4.0 * PI_D) * ratio);
      const float q = Nlm * P[l][m][j];
      Q[(l * NM + (15 + m)) * NGRID + j] = q;
      Q[(l * NM + (15 - m)) * NGRID + j] = (m & 1) ? -q : q;
    }
  }
}

// ---------------------------------------------------------------------------
// Kernel 1: build fused transform matrices.
//   A1[lm][jp]  (analysis):  Qw[l,m,j] * {C2,S2}[m,p]   (quadrature folded in)
//   A2[jp][lm]  (synthesis): scale_l * Qs[l,m,j] * {C2,S2}[m,p]
// with C2 = cos(2pi*m_hat*p/n), S2 = -sin(2pi*m_hat*p/n); rows with m==31 are 0.
// ---------------------------------------------------------------------------
__global__ void sph_buildmats_kernel(float* __restrict__ ws) {
  const int idx = blockIdx.x * blockDim.x + threadIdx.x;
  if (idx >= KLM * KJP) return;
  const int lm = idx >> 10, jp = idx & 1023;
  const int l = lm >> 5, m = lm & 31;
  const int jj = jp >> 5, p = jp & 31;

  float a1r = 0.f, a1i = 0.f, a2r = 0.f, a2i = 0.f;
  if (m < NM) {
    const float q = ws[WS_Q + (l * NM + m) * NGRID + jj];
    const int mh = m - 15;
    const float ang = (2.0f * (float)PI_D / NGRID) * (float)(mh * p);
    float s, c;
    sincosf(ang, &s, &c);
    const float C2 = c, S2 = -s;
    const float th = (float)PI_D * (jj + 0.5f) / NGRID;
    const float wj = sinf(th) * ((float)PI_D / NGRID) * (2.0f * (float)PI_D / NGRID);
    const float scale_l = 2.0f * (float)PI_D * sqrtf(4.0f * (float)PI_D / (2.0f * l + 1.0f));
    a1r = q * wj * C2;      a1i = q * wj * S2;
    a2r = q * scale_l * C2; a2i = q * scale_l * S2;
  }
  ws[WS_A1RE + lm * KJP + jp] = a1r;
  ws[WS_A1IM + lm * KJP + jp] = a1i;
  ws[WS_A2RE + jp * KLM + lm] = a2r;
  ws[WS_A2IM + jp * KLM + lm] = a2i;
}

// ---------------------------------------------------------------------------
// Wave-level fp32 WMMA GEMM: one wave accumulates a 32x128 strip of C
// (2 M-tiles x 8 N-tiles) so every B fragment feeds two WMMAs.
// ISA layouts (CDNA5 ISA):
//   A 16x4 : lanes 0-15 M=lane, v0=K0 v1=K1; lanes 16-31 v0=K2 v1=K3
//   B 4x16 : lanes 0-15 N=lane, v0=K0 v1=K1; lanes 16-31 v0=K2 v1=K3
//   C 16x16: VGPR r = row r (lanes 0-15) / row r+8 (lanes 16-31)
// ---------------------------------------------------------------------------
__device__ __forceinline__ void wave_gemm_32x128(
    const float* __restrict__ A, int lda,
    const float* __restrict__ B, int ldb,
    int K, v8f acc[2][8]) {
  const int lane = threadIdx.x & 31;
  const int row  = lane & 15;
  const int kh   = (lane >> 4) << 1;      // 0 or 2
  for (int k = 0; k < K; k += 4) {
    v2f a0, a1;
    a0.x = A[row * lda + k + kh];
    a0.y = A[row * lda + k + kh + 1];
    a1.x = A[(row + 16) * lda + k + kh];
    a1.y = A[(row + 16) * lda + k + kh + 1];
#pragma unroll
    for (int nt = 0; nt < 8; ++nt) {
      v2f bb;
      bb.x = B[(k + kh) * ldb + nt * 16 + row];
      bb.y = B[(k + kh + 1) * ldb + nt * 16 + row];
      acc[0][nt] = __builtin_amdgcn_wmma_f32_16x16x4_f32(
          false, a0, false, bb, (short)0, acc[0][nt], false, false);
      acc[1][nt] = __builtin_amdgcn_wmma_f32_16x16x4_f32(
          false, a1, false, bb, (short)0, acc[1][nt], false, false);
    }
  }
}

__device__ __forceinline__ void wave_store_32x128(
    float* __restrict__ C, int ldc, const v8f acc[2][8]) {
  const int lane = threadIdx.x & 31;
  const int col  = lane & 15;
  const int rb   = (lane >> 4) * 8;
#pragma unroll
  for (int t = 0; t < 2; ++t)
#pragma unroll
    for (int nt = 0; nt < 8; ++nt)
#pragma unroll
      for (int v = 0; v < 8; ++v)
        C[(t * 16 + rb + v) * ldc + nt * 16 + col] = acc[t][nt][v];
}

// Phase A: coef_p[b] (512x128) = A1_p (512x1024) @ inputs[b] (1024x128)
__global__ void __launch_bounds__(32, 1)
sph_phaseA_kernel(const float* __restrict__ A1,
                  const float* __restrict__ X,
                  float* __restrict__ coef) {
  const int mt = blockIdx.x;   // 0..15 -> 32-row strip
  const int b  = blockIdx.y;   // 0..63
  const float* A = A1 + (size_t)mt * 32 * KJP;
  const float* B = X + (size_t)b * KJP * CIN;
  float* C = coef + (size_t)b * KLM * CIN + (size_t)mt * 32 * CIN;
  v8f acc[2][8] = {};
  wave_gemm_32x128(A, KJP, B, CIN, KJP, acc);
  wave_store_32x128(C, CIN, acc);
}

// Phase B: chat_p[b,l] (32x128) = coef_p[b,l] (32x128) @ weights[:, l, :]
// A 32-row strip is exactly one degree-l group (m = 0..31 padded).
__global__ void __launch_bounds__(32, 1)
sph_phaseB_kernel(const float* __restrict__ coef,
                  const float* __restrict__ W,
                  float* __restrict__ chat) {
  const int l = blockIdx.x;    // 0..15 == degree
  const int b = blockIdx.y;
  const float* A = coef + (size_t)b * KLM * CIN + (size_t)l * 32 * CIN;
  const float* B = W + (size_t)l * COUT;          // row stride = LMAX*COUT
  float* C = chat + (size_t)b * KLM * COUT + (size_t)l * 32 * COUT;
  v8f acc[2][8] = {};
  wave_gemm_32x128(A, CIN, B, LMAX * COUT, CIN, acc);
  wave_store_32x128(C, COUT, acc);
}

// Phase C: out[b] (1024x128) = A2re @ chat_re[b] + A2im @ chat_im[b] + bias
__global__ void __launch_bounds__(32, 1)
sph_phaseC_kernel(const float* __restrict__ A2re,
                  const float* __restrict__ A2im,
                  const float* __restrict__ chre,
                  const float* __restrict__ chim,
                  const float* __restrict__ bias,
                  float* __restrict__ out) {
  const int mt = blockIdx.x;   // 0..31 -> 32 jp rows
  const int b  = blockIdx.y;
  const int lane = threadIdx.x & 31;
  v8f acc[2][8];
#pragma unroll
  for (int nt = 0; nt < 8; ++nt) {
    const float bv = bias[nt * 16 + (lane & 15)];
#pragma unroll
    for (int v = 0; v < 8; ++v) {      // whole tile column shares bias[o]
      acc[0][nt][v] = bv;
      acc[1][nt][v] = bv;
    }
  }
  const float* Cre = chre + (size_t)b * KLM * COUT;
  const float* Cim = chim + (size_t)b * KLM * COUT;
  wave_gemm_32x128(A2re + (size_t)mt * 32 * KLM, KLM, Cre, COUT, KLM, acc);
  wave_gemm_32x128(A2im + (size_t)mt * 32 * KLM, KLM, Cim, COUT, KLM, acc);
  wave_store_32x128(out + (size_t)b * KJP * COUT + (size_t)mt * 32 * COUT, COUT, acc);
}

extern "C" void kernel_launch(void* const* d_in, const int* in_sizes, int n_in,
                              void* d_out, int out_size, void* d_ws, size_t ws_size,
                              hipStream_t stream) {
  const float* inp  = (const float*)d_in[0];   // (64,32,32,128)
  const float* wgt  = (const float*)d_in[1];   // (128,16,128)
  const float* bias = (const float*)d_in[2];   // (128)
  float* out = (float*)d_out;                  // (64,32,32,128)
  float* ws  = (float*)d_ws;                   // ~76 MB used

  sph_legendre_kernel<<<1, 32, 0, stream>>>(ws);
  sph_buildmats_kernel<<<(KLM * KJP + 255) / 256, 256, 0, stream>>>(ws);

  dim3 gAB(16, BATCH);
  sph_phaseA_kernel<<<gAB, 32, 0, stream>>>(ws + WS_A1RE, inp, ws + WS_COEFRE);
  sph_phaseA_kernel<<<gAB, 32, 0, stream>>>(ws + WS_A1IM, inp, ws + WS_COEFIM);
  sph_phaseB_kernel<<<gAB, 32, 0, stream>>>(ws + WS_COEFRE, wgt, ws + WS_CHATRE);
  sph_phaseB_kernel<<<gAB, 32, 0, stream>>>(ws + WS_COEFIM, wgt, ws + WS_CHATIM);

  dim3 gC(32, BATCH);
  sph_phaseC_kernel<<<gC, 32, 0, stream>>>(ws + WS_A2RE, ws + WS_A2IM,
                                           ws + WS_CHATRE, ws + WS_CHATIM,
                                           bias, out);
}